// EnergyOut_44057774522749
// MI455X (gfx1250) — compile-verified
//
#include <hip/hip_runtime.h>

typedef __attribute__((ext_vector_type(2))) float v2f;
typedef __attribute__((ext_vector_type(8))) float v8f;

constexpr int kNodes   = 500000;
constexpr int kNodeDim = 128;
constexpr int kGraphs  = 16384;
constexpr int kTiles   = kNodes / 16;   // 31250, exact

// One wave32 per 16-node tile: [16x128] x [128x64] via V_WMMA_F32_16X16X4_F32,
// then bias + SiLU + 64->1 dot (W2) reduced with shfl_xor butterflies.
__global__ void __launch_bounds__(256) node_mlp_kernel(
    const float* __restrict__ X,     // [N,128]
    const float* __restrict__ aein,  // [N]
    const float* __restrict__ W1,    // [64,128]
    const float* __restrict__ b1,    // [64]
    const float* __restrict__ W2,    // [1,64]
    const float* __restrict__ b2,    // [1]
    float* __restrict__ aout)        // [N]
{
    const int lane = threadIdx.x & 31;
    const int wave = threadIdx.x >> 5;
    const int j0   = lane & 15;      // matrix row/col within tile
    const int half = lane >> 4;      // K half: lanes 0-15 -> K{0,1}, 16-31 -> K{2,3}

    int tile = blockIdx.x * 8 + wave;
    const bool ok = (tile < kTiles);
    if (!ok) tile = kTiles - 1;      // clamp: keep EXEC all-1s for WMMA, predicate stores
    const int nodeBase = tile * 16;

    const float* arow  = X  + (size_t)(nodeBase + j0) * kNodeDim + 2 * half;
    const float* brow0 = W1 + (size_t)( 0 + j0) * kNodeDim + 2 * half;
    const float* brow1 = W1 + (size_t)(16 + j0) * kNodeDim + 2 * half;
    const float* brow2 = W1 + (size_t)(32 + j0) * kNodeDim + 2 * half;
    const float* brow3 = W1 + (size_t)(48 + j0) * kNodeDim + 2 * half;

    v8f acc0 = {}, acc1 = {}, acc2 = {}, acc3 = {};

#pragma unroll 4
    for (int k = 0; k < kNodeDim; k += 4) {
        // A: streamed once -> non-temporal. B: 32KB W1, stays hot in WGP$/L2.
        v2f a  = __builtin_nontemporal_load((const v2f*)(arow + k));
        v2f B0 = *(const v2f*)(brow0 + k);
        v2f B1 = *(const v2f*)(brow1 + k);
        v2f B2 = *(const v2f*)(brow2 + k);
        v2f B3 = *(const v2f*)(brow3 + k);
        acc0 = __builtin_amdgcn_wmma_f32_16x16x4_f32(false, a, false, B0, (short)0, acc0, false, false);
        acc1 = __builtin_amdgcn_wmma_f32_16x16x4_f32(false, a, false, B1, (short)0, acc1, false, false);
        acc2 = __builtin_amdgcn_wmma_f32_16x16x4_f32(false, a, false, B2, (short)0, acc2, false, false);
        acc3 = __builtin_amdgcn_wmma_f32_16x16x4_f32(false, a, false, B3, (short)0, acc3, false, false);
    }

    // Per-lane epilogue constants (n = 16t + j0, identical for both halves)
    const float b1v0 = b1[j0], b1v1 = b1[16 + j0], b1v2 = b1[32 + j0], b1v3 = b1[48 + j0];
    const float w2v0 = W2[j0], w2v1 = W2[16 + j0], w2v2 = W2[32 + j0], w2v3 = W2[48 + j0];
    const float b2s  = b2[0];

    // acc_t[r] = preact h[M = r + 8*half][N = 16t + j0]
    float p[8];
#pragma unroll
    for (int r = 0; r < 8; ++r) {
        float h0 = acc0[r] + b1v0;
        float h1 = acc1[r] + b1v1;
        float h2 = acc2[r] + b1v2;
        float h3 = acc3[r] + b1v3;
        h0 = h0 / (1.0f + __expf(-h0));   // SiLU
        h1 = h1 / (1.0f + __expf(-h1));
        h2 = h2 / (1.0f + __expf(-h2));
        h3 = h3 / (1.0f + __expf(-h3));
        p[r] = h0 * w2v0 + h1 * w2v1 + h2 * w2v2 + h3 * w2v3;
    }

    // Reduce across the 16 lanes of each half (masks < 16 stay within a half)
#pragma unroll
    for (int m = 1; m < 16; m <<= 1) {
#pragma unroll
        for (int r = 0; r < 8; ++r) p[r] += __shfl_xor(p[r], m, 32);
    }

    // Lane L in {0..7} writes row L; lane 16+L writes row 8+L.
    const int rr = lane & 7;
    float v = p[0];
    v = (rr == 1) ? p[1] : v;
    v = (rr == 2) ? p[2] : v;
    v = (rr == 3) ? p[3] : v;
    v = (rr == 4) ? p[4] : v;
    v = (rr == 5) ? p[5] : v;
    v = (rr == 6) ? p[6] : v;
    v = (rr == 7) ? p[7] : v;

    if (ok && j0 < 8) {
        const int node = nodeBase + rr + 8 * half;
        aout[node] = aein[node] + v + b2s;
    }
}

// Deterministic segment sum: batch is sorted, one thread per graph,
// binary-search segment bounds then serial accumulate. No atomics.
__global__ void __launch_bounds__(256) segsum_kernel(
    const float* __restrict__ ae,     // [N] atomic energies (written above)
    const int*   __restrict__ batch,  // [N] sorted graph ids
    float* __restrict__ tot)          // [G]
{
    const int g = blockIdx.x * blockDim.x + threadIdx.x;
    if (g >= kGraphs) return;

    int lo = 0, hi = kNodes;              // lower_bound(g)
    while (lo < hi) { int mid = (lo + hi) >> 1; if (batch[mid] <  g) lo = mid + 1; else hi = mid; }
    const int s = lo;
    hi = kNodes;                          // lower_bound(g+1)
    while (lo < hi) { int mid = (lo + hi) >> 1; if (batch[mid] <= g) lo = mid + 1; else hi = mid; }
    const int e = lo;

    float sum = 0.0f;
    for (int i = s; i < e; ++i) sum += ae[i];
    tot[g] = sum;
}

extern "C" void kernel_launch(void* const* d_in, const int* in_sizes, int n_in,
                              void* d_out, int out_size, void* d_ws, size_t ws_size,
                              hipStream_t stream) {
    const float* X    = (const float*)d_in[0];  // node_scalar [500000,128]
    const float* aein = (const float*)d_in[1];  // atomic_energies_in [500000]
    const int*   batch= (const int*)  d_in[2];  // batch [500000] (sorted)
    const float* W1   = (const float*)d_in[3];  // [64,128]
    const float* b1   = (const float*)d_in[4];  // [64]
    const float* W2   = (const float*)d_in[5];  // [1,64]
    const float* b2   = (const float*)d_in[6];  // [1]

    float* aout = (float*)d_out;                // [500000]
    float* tot  = aout + kNodes;                // [16384]

    const int blocks = (kTiles + 7) / 8;        // 8 waves (tiles) per 256-thread block
    node_mlp_kernel<<<blocks, 256, 0, stream>>>(X, aein, W1, b1, W2, b2, aout);
    segsum_kernel<<<(kGraphs + 255) / 256, 256, 0, stream>>>(aout, batch, tot);
}